// RVQ_60146722013380
// MI455X (gfx1250) — compile-verified
//
#include <hip/hip_runtime.h>

// ---------------------------------------------------------------------------
// RVQ on MI455X (gfx1250): fused f32 WMMA GEMM + argmin per stage.
//   x:         [16, 1024, 256] f32  -> N = 16384 rows, D = 256
//   codebooks: [8, 1024, 256]  f32  -> 8 stages, 1024 codes each
// d_out layout (f32): [loss_cd, loss_enc, enc(16384*8), quantised(16384*256)]
// ---------------------------------------------------------------------------

typedef __attribute__((ext_vector_type(2))) float v2f;
typedef __attribute__((ext_vector_type(8))) float v8f;

#define NROWS 16384
#define D 256
#define CODES 1024
#define NSTAGES 8
#define ROWS_PER_BLOCK 128   // 8 waves x 16 rows
#define LDS_STRIDE 260       // 256 + 4: rotates LDS banks by 4 per row; 1040B row, 16B aligned

// ---------------- init: res = x, rowloss = 0 ----------------
__global__ __launch_bounds__(256) void rvq_init_kernel(const float* __restrict__ x,
                                                       float* __restrict__ res,
                                                       float* __restrict__ rowloss) {
    int idx = blockIdx.x * 256 + threadIdx.x;           // over 1,048,576 float4
    const float4* xv = (const float4*)x;
    float4* rv = (float4*)res;
    rv[idx] = xv[idx];
    if (idx < NROWS) rowloss[idx] = 0.0f;
}

// ---------------- c2[j] = ||codebook row j||^2, all 8192 codes ----------------
__global__ __launch_bounds__(256) void rvq_c2_kernel(const float* __restrict__ cb,
                                                     float* __restrict__ c2) {
    int j = blockIdx.x * 256 + threadIdx.x;             // 0 .. 8191
    const float4* row = (const float4*)(cb + (size_t)j * D);
    float s = 0.0f;
    #pragma unroll 8
    for (int i = 0; i < D / 4; ++i) {
        float4 v = row[i];
        s += v.x * v.x + v.y * v.y + v.z * v.z + v.w * v.w;
    }
    c2[j] = s;
}

// ---------------- per-stage fused GEMM (WMMA f32) + argmin ----------------
// Each wave: 16 rows. Block: 8 waves = 128 rows. Sweep 64 column tiles of 16 codes.
__global__ __launch_bounds__(256) void rvq_argmin_kernel(const float* __restrict__ res,
                                                         const float* __restrict__ cb,   // stage codebook [1024, 256]
                                                         const float* __restrict__ c2,   // stage norms [1024]
                                                         int* __restrict__ enc_out,      // [N]
                                                         float* __restrict__ enc_f_out,  // [N, 8]
                                                         int stage) {
    extern __shared__ float lds[];
    float* resT = lds;                                   // [128][LDS_STRIDE]
    float* cbT  = lds + ROWS_PER_BLOCK * LDS_STRIDE;     // [16][LDS_STRIDE]

    const int t    = threadIdx.x;
    const int lane = t & 31;
    const int wave = t >> 5;
    const int l16  = lane & 15;   // col (B/C) or row (A) within tile
    const int hi   = lane >> 4;   // selects K pair (A/B) / upper M half (C)
    const int row0 = blockIdx.x * ROWS_PER_BLOCK;

    // Stage residual tile into LDS (float4 global loads, float4 LDS stores).
    {
        const float4* src = (const float4*)(res + (size_t)row0 * D);
        for (int i = t; i < ROWS_PER_BLOCK * (D / 4); i += 256) {
            int r = i >> 6, kc = i & 63;
            *(float4*)(resT + r * LDS_STRIDE + kc * 4) = src[r * 64 + kc];
        }
    }

    float best[8];
    int   bidx[8];
    #pragma unroll
    for (int r = 0; r < 8; ++r) { best[r] = 3.4028235e38f; bidx[r] = 0; }

    // A-fragment base for this lane: row (wave*16 + l16), K pair selected by hi.
    const float* aBase = resT + (size_t)(wave * 16 + l16) * LDS_STRIDE + hi * 2;
    const float* bBase = cbT  + (size_t)l16 * LDS_STRIDE + hi * 2;

    for (int ct = 0; ct < CODES / 16; ++ct) {
        __syncthreads();                                  // protect cbT reuse
        // Cooperative load of 16-code B tile.
        const float4* src = (const float4*)(cb + (size_t)ct * 16 * D);
        for (int i = t; i < 16 * (D / 4); i += 256) {
            int r = i >> 6, kc = i & 63;
            *(float4*)(cbT + r * LDS_STRIDE + kc * 4) = src[r * 64 + kc];
        }
        __syncthreads();

        v8f c = {0.f, 0.f, 0.f, 0.f, 0.f, 0.f, 0.f, 0.f};
        #pragma unroll 8
        for (int k = 0; k < D; k += 4) {
            v2f a = *(const v2f*)(aBase + k);             // ds_load_b64, conflict-free
            v2f b = *(const v2f*)(bBase + k);
            // D(16x16 f32) = A(16x4 f32) * B(4x16 f32) + C
            c = __builtin_amdgcn_wmma_f32_16x16x4_f32(
                    /*neg_a=*/false, a, /*neg_b=*/false, b,
                    /*c_mod=*/(short)0, c, /*reuse_a=*/false, /*reuse_b=*/false);
        }

        // score(row, col) = c2[col] - 2*dot  (row-constant ||res||^2 dropped)
        const int   col = ct * 16 + l16;
        const float cc  = c2[col];
        #pragma unroll
        for (int r = 0; r < 8; ++r) {
            float s = cc - 2.0f * c[r];
            if (s < best[r]) { best[r] = s; bidx[r] = col; }  // strict < keeps earliest col
        }
    }

    // Min-reduce across the 16 lanes holding the same row; tie -> smaller index.
    #pragma unroll
    for (int m = 8; m >= 1; m >>= 1) {
        #pragma unroll
        for (int r = 0; r < 8; ++r) {
            float os = __shfl_xor(best[r], m, 16);
            int   oi = __shfl_xor(bidx[r], m, 16);
            if (os < best[r] || (os == best[r] && oi < bidx[r])) { best[r] = os; bidx[r] = oi; }
        }
    }

    if (l16 == 0) {   // lane 0 -> rows M=0..7, lane 16 -> rows M=8..15
        #pragma unroll
        for (int r = 0; r < 8; ++r) {
            int m = r + 8 * hi;
            int n = row0 + wave * 16 + m;
            enc_out[n] = bidx[r];
            enc_f_out[(size_t)n * NSTAGES + stage] = (float)bidx[r];
        }
    }
}

// ---------------- per-stage residual update + per-row loss ----------------
// One wave per row; rowloss[n] += ||res - cb[enc[n]]||^2  (== min squared distance)
__global__ __launch_bounds__(256) void rvq_update_kernel(float* __restrict__ res,
                                                         const float* __restrict__ cb,
                                                         const int* __restrict__ enc,
                                                         float* __restrict__ rowloss) {
    const int wave = threadIdx.x >> 5;
    const int lane = threadIdx.x & 31;
    const int n = blockIdx.x * 8 + wave;

    const float4* code = (const float4*)(cb + (size_t)enc[n] * D);
    float4* r4 = (float4*)(res + (size_t)n * D);

    float ss = 0.0f;
    #pragma unroll
    for (int i = lane; i < D / 4; i += 32) {
        float4 rv = r4[i], cv = code[i];
        rv.x -= cv.x; rv.y -= cv.y; rv.z -= cv.z; rv.w -= cv.w;
        ss += rv.x * rv.x + rv.y * rv.y + rv.z * rv.z + rv.w * rv.w;
        r4[i] = rv;
    }
    #pragma unroll
    for (int m = 16; m >= 1; m >>= 1) ss += __shfl_xor(ss, m, 32);
    if (lane == 0) rowloss[n] += ss;   // single writer per row, stages serialized
}

// ---------------- deterministic loss reduction ----------------
__global__ __launch_bounds__(256) void rvq_loss_kernel(const float* __restrict__ rowloss,
                                                       float* __restrict__ out) {
    __shared__ float sm[256];
    const int t = threadIdx.x;
    float s = 0.0f;
    for (int n = t; n < NROWS; n += 256) s += rowloss[n];   // fixed order per slot
    sm[t] = s;
    __syncthreads();
    for (int w = 128; w > 0; w >>= 1) {
        if (t < w) sm[t] += sm[t + w];
        __syncthreads();
    }
    if (t == 0) {
        float loss = sm[0] / (float)((size_t)NROWS * D);
        out[0] = loss;   // loss_cd
        out[1] = loss;   // loss_enc (same forward value)
    }
}

// ---------------- quantised = x - res_final (float2: dest is 8B aligned) ----------------
__global__ __launch_bounds__(256) void rvq_quant_kernel(const float* __restrict__ x,
                                                        const float* __restrict__ res,
                                                        float* __restrict__ q) {
    int idx = blockIdx.x * 256 + threadIdx.x;              // over 2,097,152 float2
    const float2* xv = (const float2*)x;
    const float2* rv = (const float2*)res;
    float2 v = xv[idx], r = rv[idx];
    v.x -= r.x; v.y -= r.y;
    ((float2*)q)[idx] = v;
}

// ---------------------------------------------------------------------------
extern "C" void kernel_launch(void* const* d_in, const int* in_sizes, int n_in,
                              void* d_out, int out_size, void* d_ws, size_t ws_size,
                              hipStream_t stream) {
    const float* x  = (const float*)d_in[0];   // [16,1024,256]
    const float* cb = (const float*)d_in[1];   // [8,1024,256]
    float* out = (float*)d_out;

    // Workspace layout (floats)
    float* ws      = (float*)d_ws;
    float* res     = ws;                                  // 4,194,304
    float* rowloss = ws + (size_t)NROWS * D;              // 16,384
    float* c2      = rowloss + NROWS;                     // 8,192
    int*   enc     = (int*)(c2 + NSTAGES * CODES);        // 16,384

    float* enc_f = out + 2;                               // [N, 8]
    float* quant = out + 2 + (size_t)NROWS * NSTAGES;     // [N, 256]

    rvq_init_kernel<<<(NROWS * D / 4) / 256, 256, 0, stream>>>(x, res, rowloss);
    rvq_c2_kernel<<<(NSTAGES * CODES) / 256, 256, 0, stream>>>(cb, c2);

    const size_t shmem = (size_t)(ROWS_PER_BLOCK + 16) * LDS_STRIDE * sizeof(float); // ~146 KB
    for (int s = 0; s < NSTAGES; ++s) {
        const float* cbS = cb + (size_t)s * CODES * D;
        rvq_argmin_kernel<<<NROWS / ROWS_PER_BLOCK, 256, shmem, stream>>>(
            res, cbS, c2 + (size_t)s * CODES, enc, enc_f, s);
        rvq_update_kernel<<<NROWS / 8, 256, 0, stream>>>(res, cbS, enc, rowloss);
    }

    rvq_loss_kernel<<<1, 256, 0, stream>>>(rowloss, out);
    rvq_quant_kernel<<<(NROWS * D / 2) / 256, 256, 0, stream>>>(x, res, quant);
}